// NextIntegerEncoder_15522011808326
// MI455X (gfx1250) — compile-verified
//
#include <hip/hip_runtime.h>

typedef float v2f __attribute__((ext_vector_type(2)));
typedef float v8f __attribute__((ext_vector_type(8)));

constexpr int NNODES = 100000;
constexpr int F = 64;          // hidden width (and input width)
constexpr int EMB = 32;        // final embedding width

// Hardware fp32 atomic add (global_atomic_add_f32), never a CAS loop.
__device__ __forceinline__ void atomic_add_f32(float* p, float v) {
    unsafeAtomicAdd(p, v);
}

// ---------------- elementwise helpers ----------------

__global__ void zero_kernel(float* __restrict__ p, int n) {
    int i = blockIdx.x * blockDim.x + threadIdx.x;
    if (i < n) p[i] = 0.0f;
}

__global__ void degree_kernel(const int* __restrict__ dst, int E,
                              float* __restrict__ deg) {
    int e = blockIdx.x * blockDim.x + threadIdx.x;
    if (e < E) atomic_add_f32(&deg[dst[e]], 1.0f);
}

__global__ void dinv_kernel(float* __restrict__ deg, int n) {
    int i = blockIdx.x * blockDim.x + threadIdx.x;
    if (i < n) deg[i] = rsqrtf(deg[i] + 1.0f);   // deg includes self-loop (+1)
}

// agg = relu(agg + h*dinv^2 + b)   (in place)
__global__ void combine_relu_kernel(const float* __restrict__ h,
                                    const float* __restrict__ bias,
                                    const float* __restrict__ dinv,
                                    float* __restrict__ agg, int n) {
    int i = blockIdx.x * blockDim.x + threadIdx.x;
    if (i < n) {
        int node = i >> 6;           // /64
        int f    = i & 63;
        float di = dinv[node];
        float v  = agg[i] + h[i] * di * di + bias[f];
        agg[i] = fmaxf(v, 0.0f);
    }
}

// ---------------- edge scatter (dominant cost) ----------------
// 16 threads per edge, each handles a float4 slab of the 64 features.
// Consecutive threads of a wave read contiguous 256B of h[src] (coalesced,
// mostly L2-resident: h is 25.6MB vs 192MB L2); the adds are single
// global_atomic_add_f32 instructions resolved at L2.
__global__ void scatter_kernel(const int* __restrict__ src,
                               const int* __restrict__ dst, int E,
                               const float* __restrict__ h,
                               const float* __restrict__ dinv,
                               float* __restrict__ agg) {
    long long idx = (long long)blockIdx.x * blockDim.x + threadIdx.x;
    long long total = (long long)E * 16;
    if (idx >= total) return;
    int e  = (int)(idx >> 4);
    int fg = ((int)idx & 15) * 4;
    int s = src[e];
    int d = dst[e];
    float w = dinv[s] * dinv[d];
    const float4 hv = *(const float4*)(h + (long long)s * F + fg);
    float* ap = agg + (long long)d * F + fg;
    atomic_add_f32(ap + 0, hv.x * w);
    atomic_add_f32(ap + 1, hv.y * w);
    atomic_add_f32(ap + 2, hv.z * w);
    atomic_add_f32(ap + 3, hv.w * w);
}

// ---------------- WMMA fp32 GEMM: H[NNODES x NOUT] = X[NNODES x 64] @ W[64 x NOUT] ----------------
// One wave per 16-row stripe; NOUT/16 accumulator tiles so the A fragment is
// reused across all N tiles. K stepped by 4 via V_WMMA_F32_16X16X4_F32.
// Fragment layouts per CDNA5 ISA 7.12.2 (32-bit A 16x4, 32-bit C/D 16x16).
template <int NOUT, bool ADD_BIAS>
__global__ __launch_bounds__(32) void gemm_wmma_f32(const float* __restrict__ X,
                                                    const float* __restrict__ W,
                                                    const float* __restrict__ bias,
                                                    float* __restrict__ H) {
    const int lane  = threadIdx.x;        // 0..31, EXEC all-ones (no divergence)
    const int tm    = blockIdx.x;         // row tile (6250 tiles, exact)
    const int mlo   = lane & 15;
    const int hi    = lane >> 4;          // 0: K=0,1 | 1: K=2,3
    const int khalf = hi * 2;

    constexpr int NT = NOUT / 16;
    v8f acc[NT] = {};

    const float* xrow = X + (long long)(tm * 16 + mlo) * F;

    #pragma unroll
    for (int k = 0; k < F; k += 4) {
        v2f a;
        a.x = xrow[k + khalf];
        a.y = xrow[k + khalf + 1];
        #pragma unroll
        for (int t = 0; t < NT; ++t) {
            v2f b;
            int n = t * 16 + mlo;
            b.x = W[(k + khalf) * NOUT + n];
            b.y = W[(k + khalf + 1) * NOUT + n];
            // 8 args: (neg_a, A, neg_b, B, c_mod, C, reuse_a, reuse_b)
            acc[t] = __builtin_amdgcn_wmma_f32_16x16x4_f32(
                false, a, false, b, (short)0, acc[t], false, false);
        }
    }

    // C/D layout: VGPR r -> (M = r + 8*hi, N = lane&15) within the tile
    #pragma unroll
    for (int t = 0; t < NT; ++t) {
        int nn = t * 16 + mlo;
        float badd = ADD_BIAS ? bias[nn] : 0.0f;
        #pragma unroll
        for (int r = 0; r < 8; ++r) {
            int mm = tm * 16 + r + hi * 8;
            H[(long long)mm * NOUT + nn] = acc[t][r] + badd;
        }
    }
}

// ---------------- driver ----------------

extern "C" void kernel_launch(void* const* d_in, const int* in_sizes, int n_in,
                              void* d_out, int out_size, void* d_ws, size_t ws_size,
                              hipStream_t stream) {
    const float* x   = (const float*)d_in[0];
    const int*   ei  = (const int*)d_in[1];
    const float* W1  = (const float*)d_in[2];
    const float* b1  = (const float*)d_in[3];
    const float* W2  = (const float*)d_in[4];
    const float* b2  = (const float*)d_in[5];
    const float* Wf  = (const float*)d_in[6];
    const float* bf  = (const float*)d_in[7];
    float* out = (float*)d_out;

    const int E = in_sizes[1] / 2;
    const int* src = ei;
    const int* dst = ei + E;

    // workspace layout (floats): dinv[NNODES] | hA[NNODES*64] | hB[NNODES*64]
    float* dinv = (float*)d_ws;
    float* hA   = dinv + NNODES;
    float* hB   = hA + (size_t)NNODES * F;

    const int TB = 256;
    const int nNF = NNODES * F;
    dim3 blk(TB);
    dim3 grid_nodes((NNODES + TB - 1) / TB);
    dim3 grid_nf((nNF + TB - 1) / TB);
    dim3 grid_edges((E + TB - 1) / TB);
    dim3 grid_scatter((int)(((long long)E * 16 + TB - 1) / TB));
    dim3 grid_gemm(NNODES / 16);   // 6250, exact
    dim3 wave(32);

    // degrees -> dinv (in place)
    zero_kernel<<<grid_nodes, blk, 0, stream>>>(dinv, NNODES);
    degree_kernel<<<grid_edges, blk, 0, stream>>>(dst, E, dinv);
    dinv_kernel<<<grid_nodes, blk, 0, stream>>>(dinv, NNODES);

    // ---- layer 1: hB = relu(scatter(x@W1) + (x@W1)*dinv^2 + b1) ----
    gemm_wmma_f32<F, false><<<grid_gemm, wave, 0, stream>>>(x, W1, nullptr, hA);
    zero_kernel<<<grid_nf, blk, 0, stream>>>(hB, nNF);
    scatter_kernel<<<grid_scatter, blk, 0, stream>>>(src, dst, E, hA, dinv, hB);
    combine_relu_kernel<<<grid_nf, blk, 0, stream>>>(hA, b1, dinv, hB, nNF);

    // ---- layer 2: hB = relu(scatter(h1@W2) + (h1@W2)*dinv^2 + b2) ----
    gemm_wmma_f32<F, false><<<grid_gemm, wave, 0, stream>>>(hB, W2, nullptr, hA);
    zero_kernel<<<grid_nf, blk, 0, stream>>>(hB, nNF);
    scatter_kernel<<<grid_scatter, blk, 0, stream>>>(src, dst, E, hA, dinv, hB);
    combine_relu_kernel<<<grid_nf, blk, 0, stream>>>(hA, b2, dinv, hB, nNF);

    // ---- final projection: out = h2 @ Wf + bf ----
    gemm_wmma_f32<EMB, true><<<grid_gemm, wave, 0, stream>>>(hB, Wf, bf, out);
}